// Discriminator_suvey_test3_58609123721853
// MI455X (gfx1250) — compile-verified
//
#include <hip/hip_runtime.h>
#include <hip/hip_bf16.h>
#include <math.h>
#include <stdint.h>

typedef __attribute__((ext_vector_type(2))) float v2f;
typedef __attribute__((ext_vector_type(8))) float v8f;

#define LCONST 2048
#define BCONST 4

// ---------------------------------------------------------------------------
// pack h0 = concat(x [B,80,L], repeat(enc^T,2) [B,256,L]) -> [B,336,L]
// ---------------------------------------------------------------------------
__global__ __launch_bounds__(256) void pack_h0_kernel(
    const float* __restrict__ x, const float* __restrict__ enc,
    float* __restrict__ h0)
{
    long i = (long)blockIdx.x * 256 + threadIdx.x;
    const long total = (long)BCONST * 336 * LCONST;
    if (i >= total) return;
    int l = (int)(i & (LCONST - 1));
    long t = i >> 11;             // / 2048
    int c = (int)(t % 336);
    int b = (int)(t / 336);
    float v;
    if (c < 80) {
        v = x[((long)b * 80 + c) * LCONST + l];
    } else {
        // enc: [B, L/2, 256]; nearest upsample x2 in time
        v = enc[((long)b * (LCONST / 2) + (l >> 1)) * 256 + (c - 80)];
    }
    h0[i] = v;
}

// ---------------------------------------------------------------------------
// spectral norm: one power iteration.  sigma = ||W v||^2 / (||W v|| + eps),
// v = W^T u / (||W^T u|| + eps).  Writes 1/sigma.
// ---------------------------------------------------------------------------
__global__ __launch_bounds__(256) void specnorm_kernel(
    const float* __restrict__ W1, const float* __restrict__ u1,
    const float* __restrict__ W2, const float* __restrict__ u2,
    const float* __restrict__ W3, const float* __restrict__ u3,
    float* __restrict__ invsig)
{
    const float* W; const float* u; int M, K;
    if (blockIdx.x == 0)      { W = W1; u = u1; M = 512;  K = 336;  }
    else if (blockIdx.x == 1) { W = W2; u = u2; M = 1024; K = 512;  }
    else                      { W = W3; u = u3; M = 2048; K = 1024; }

    __shared__ float v[1024];
    __shared__ float red[256];
    const int tid = threadIdx.x;

    // t = W^T u
    for (int j = tid; j < K; j += 256) {
        float s = 0.f;
        for (int i = 0; i < M; ++i) s += W[(long)i * K + j] * u[i];
        v[j] = s;
    }
    __syncthreads();
    float ss = 0.f;
    for (int j = tid; j < K; j += 256) ss += v[j] * v[j];
    red[tid] = ss; __syncthreads();
    for (int s = 128; s > 0; s >>= 1) {
        if (tid < s) red[tid] += red[tid + s];
        __syncthreads();
    }
    float invn = 1.0f / (sqrtf(red[0]) + 1e-12f);
    __syncthreads();
    for (int j = tid; j < K; j += 256) v[j] *= invn;
    __syncthreads();

    // ||W v||^2
    float n2 = 0.f;
    for (int i = tid; i < M; i += 256) {
        float s = 0.f;
        for (int j = 0; j < K; ++j) s += W[(long)i * K + j] * v[j];
        n2 += s * s;
    }
    red[tid] = n2; __syncthreads();
    for (int s = 128; s > 0; s >>= 1) {
        if (tid < s) red[tid] += red[tid + s];
        __syncthreads();
    }
    if (tid == 0) {
        float nn  = red[0];
        float nrm = sqrtf(nn);
        float sigma = nn / (nrm + 1e-12f);
        invsig[blockIdx.x] = 1.0f / sigma;
    }
}

// ---------------------------------------------------------------------------
// Generic strided GEMM on the fp32 WMMA pipe: C[b][m][n] = sum_k A(m,k)*B(k,n)
//   A element (m,k) at Ab[m*a_rs + k*a_cs], Bb likewise.
// Block = 128 threads = 4 waves; block tile 64(M) x 128(N); wave owns a 16x128
// strip (8 accumulators, 32 WMMAs per K-stage).  K staged in 16-chunks,
// double-buffered through LDS via gfx1250 async global->LDS copies (ASYNCcnt).
// B tile stored transposed (n-major, stride 20) so each B fragment is one
// aligned ds_load_b64; A fragment is an adjacent-pair ds_load_2addr_b32.
// Epilogue: *invsigma, +bias[m], leaky-relu(0.1), or causal+pad mask (-inf).
// ---------------------------------------------------------------------------
__global__ __launch_bounds__(128) void gemm_wmma_f32(
    const float* __restrict__ A, long aBS, int a_rs, int a_cs,
    const float* __restrict__ Bm, long bBS, int b_rs, int b_cs,
    float* __restrict__ C, long cBS,
    int M, int N, int K,
    const float* __restrict__ bias,
    const float* __restrict__ invs,
    int act, const int* __restrict__ dlen, int maskmode)
{
    __shared__ float sA[2][64][17];    // 64(M) x 16(K) (+1 pad), double buffered
    __shared__ float sBt[2][128][20];  // 128(N) x 16(K) (+4 pad), transposed

    const int tid  = threadIdx.x;
    const int wave = tid >> 5;
    const int lane = tid & 31;
    const int half = lane >> 4;
    const int l15  = lane & 15;
    const int b    = blockIdx.z;
    const int m0   = blockIdx.y * 64;
    const int n0b  = blockIdx.x * 128;

    const uint64_t aBase = (uint64_t)(uintptr_t)(A + (long)b * aBS);
    const uint64_t bBase = (uint64_t)(uintptr_t)(Bm + (long)b * bBS);

    // async-stage one K-chunk (16 deep) into LDS buffer `buf`
    auto stage_async = [&](int kb, int buf) {
        // A tile: 64 x 16 (row-major)
        for (int t = tid; t < 1024; t += 128) {
            int r = t >> 4, c = t & 15;
            uint32_t lds = (uint32_t)(uintptr_t)&sA[buf][r][c];
            uint32_t off = (uint32_t)((((long)(m0 + r) * a_rs +
                                        (long)(kb + c) * a_cs)) * sizeof(float));
            asm volatile("global_load_async_to_lds_b32 %0, %1, %2"
                         :: "v"(lds), "v"(off), "s"(aBase) : "memory");
        }
        // B tile: 16(K) x 128(N), written transposed into LDS
        for (int t = tid; t < 2048; t += 128) {
            int r = t >> 7, c = t & 127;
            uint32_t lds = (uint32_t)(uintptr_t)&sBt[buf][c][r];
            uint32_t off = (uint32_t)((((long)(kb + r) * b_rs +
                                        (long)(n0b + c) * b_cs)) * sizeof(float));
            asm volatile("global_load_async_to_lds_b32 %0, %1, %2"
                         :: "v"(lds), "v"(off), "s"(bBase) : "memory");
        }
    };

    v8f acc[8] = {};
    const int mrow = wave * 16 + l15;     // this lane's A row within block tile

    int cur = 0;
    stage_async(0, 0);
    asm volatile("s_wait_asynccnt 0x0" ::: "memory");
    __syncthreads();

    for (int kb = 0; kb < K; kb += 16) {
        if (kb + 16 < K) stage_async(kb + 16, cur ^ 1);

#pragma unroll
        for (int kk = 0; kk < 16; kk += 4) {
            v2f a;
            // A 16x4 fragment: VGPR0 = K{0 | 2}, VGPR1 = K{1 | 3} per half-wave
            a.x = sA[cur][mrow][kk + 2 * half];
            a.y = sA[cur][mrow][kk + 2 * half + 1];
#pragma unroll
            for (int sub = 0; sub < 8; ++sub) {
                // B 4x16 fragment, adjacent pair in transposed LDS tile
                v2f bb = *(const v2f*)&sBt[cur][sub * 16 + l15][kk + 2 * half];
                acc[sub] = __builtin_amdgcn_wmma_f32_16x16x4_f32(
                    false, a, false, bb, (short)0, acc[sub], false, false);
            }
        }

        asm volatile("s_wait_asynccnt 0x0" ::: "memory");
        __syncthreads();
        cur ^= 1;
    }

    // epilogue; C/D layout: VGPR r -> M = r + 8*half (within 16), N = l15
    const float is = invs ? invs[0] : 1.0f;
    const int dl = (dlen && maskmode) ? dlen[b] : 0;
#pragma unroll
    for (int sub = 0; sub < 8; ++sub) {
        const int n = n0b + sub * 16 + l15;
#pragma unroll
        for (int r = 0; r < 8; ++r) {
            int m = m0 + wave * 16 + r + 8 * half;
            float val = acc[sub][r] * is;
            if (bias) val += bias[m];
            if (act)  val = (val > 0.f) ? val : 0.1f * val;
            if (maskmode && (n > m || n >= dl)) val = -INFINITY;
            C[(long)b * cBS + (long)m * N + n] = val;
        }
    }
}

// ---------------------------------------------------------------------------
// in-place row softmax (rows of length Lr); one block per row
// ---------------------------------------------------------------------------
__global__ __launch_bounds__(256) void softmax_rows(float* __restrict__ p, int Lr)
{
    long row = blockIdx.x;
    float* ptr = p + row * (long)Lr;
    __shared__ float red[256];
    const int tid = threadIdx.x;

    float mx = -INFINITY;
    for (int j = tid; j < Lr; j += 256) mx = fmaxf(mx, ptr[j]);
    red[tid] = mx; __syncthreads();
    for (int s = 128; s > 0; s >>= 1) {
        if (tid < s) red[tid] = fmaxf(red[tid], red[tid + s]);
        __syncthreads();
    }
    mx = red[0]; __syncthreads();

    float sum = 0.f;
    for (int j = tid; j < Lr; j += 256) {
        float e = __expf(ptr[j] - mx);   // exp(-inf) -> 0 for masked
        ptr[j] = e;
        sum += e;
    }
    red[tid] = sum; __syncthreads();
    for (int s = 128; s > 0; s >>= 1) {
        if (tid < s) red[tid] += red[tid + s];
        __syncthreads();
    }
    float inv = 1.0f / red[0];
    for (int j = tid; j < Lr; j += 256) ptr[j] *= inv;
}

// ---------------------------------------------------------------------------
// final: o[l] = Wl . h5[:,l] + bl (zero where padded); out1[b] = mean_l o[l]
// ---------------------------------------------------------------------------
__global__ __launch_bounds__(256) void final_reduce(
    const float* __restrict__ h5, const float* __restrict__ Wl,
    const float* __restrict__ bl, const int* __restrict__ dlen,
    float* __restrict__ out1)
{
    const int b = blockIdx.x;
    const int tid = threadIdx.x;
    __shared__ float red[256];
    const float* hb = h5 + (long)b * 2048 * LCONST;
    const int dl = dlen[b];

    float acc = 0.f;
    for (int l = tid; l < LCONST; l += 256) {
        if (l < dl) {
            float s = bl[0];
            for (int c = 0; c < 2048; ++c) s += Wl[c] * hb[(long)c * LCONST + l];
            acc += s;
        }
    }
    red[tid] = acc; __syncthreads();
    for (int s = 128; s > 0; s >>= 1) {
        if (tid < s) red[tid] += red[tid + s];
        __syncthreads();
    }
    if (tid == 0) out1[b] = red[0] / (float)LCONST;
}

// ---------------------------------------------------------------------------
extern "C" void kernel_launch(void* const* d_in, const int* in_sizes, int n_in,
                              void* d_out, int out_size, void* d_ws, size_t ws_size,
                              hipStream_t stream)
{
    const float* x    = (const float*)d_in[0];
    const float* enc  = (const float*)d_in[1];
    const int*   dlen = (const int*)  d_in[2];
    // d_in[3] = max_len (known: 2048)
    const float* W1 = (const float*)d_in[4];  const float* b1 = (const float*)d_in[5];
    const float* u1 = (const float*)d_in[6];
    const float* W2 = (const float*)d_in[7];  const float* b2 = (const float*)d_in[8];
    const float* u2 = (const float*)d_in[9];
    const float* W3 = (const float*)d_in[10]; const float* b3 = (const float*)d_in[11];
    const float* u3 = (const float*)d_in[12];
    const float* Wq1 = (const float*)d_in[13]; const float* bq1 = (const float*)d_in[14];
    const float* Wk1 = (const float*)d_in[15]; const float* bk1 = (const float*)d_in[16];
    const float* Wv1 = (const float*)d_in[17]; const float* bv1 = (const float*)d_in[18];
    const float* Wq2 = (const float*)d_in[19]; const float* bq2 = (const float*)d_in[20];
    const float* Wk2 = (const float*)d_in[21]; const float* bk2 = (const float*)d_in[22];
    const float* Wv2 = (const float*)d_in[23]; const float* bv2 = (const float*)d_in[24];
    const float* Wl  = (const float*)d_in[25]; const float* bl  = (const float*)d_in[26];

    const long L = LCONST;
    const long BLL = (long)BCONST * L * L;          // 16,777,216
    float* out  = (float*)d_out;                    // out1: 4 floats
    float* p1   = out + 4;                          // [4,2048,2048]
    float* p2   = p1 + BLL;

    // workspace layout (floats)
    float* ws     = (float*)d_ws;
    float* invsig = ws;                               // 64
    float* h0     = ws + 64;                          // 4*336*2048
    float* bufA   = h0   + (long)BCONST * 336 * L;    // 64 MB (h3, then h4)
    float* bufB   = bufA + BLL;                       // 64 MB (h1, v1, h5)
    float* bufD   = bufB + BLL;                       // 64 MB (h2, v2)
    float* qbuf   = bufD + BLL;                       // 4*256*2048
    float* kbuf   = qbuf + (long)BCONST * 256 * L;

    (void)n_in; (void)in_sizes; (void)out_size; (void)ws_size;

    dim3 blk(128);

    // 1) pack input
    {
        long total = (long)BCONST * 336 * L;
        pack_h0_kernel<<<dim3((unsigned)((total + 255) / 256)), dim3(256), 0, stream>>>(x, enc, h0);
    }
    // 2) spectral-norm scales
    specnorm_kernel<<<dim3(3), dim3(256), 0, stream>>>(W1, u1, W2, u2, W3, u3, invsig);

    // 3) h1 = lrelu((W1 h0)/s1 + b1)   [512 x 2048]
    gemm_wmma_f32<<<dim3(16, 8, 4), blk, 0, stream>>>(
        W1, 0, 336, 1,  h0, 336 * L, (int)L, 1,
        bufB, 512 * L, 512, (int)L, 336, b1, invsig + 0, 1, nullptr, 0);
    // 4) h2 = lrelu((W2 h1)/s2 + b2)   [1024 x 2048]
    gemm_wmma_f32<<<dim3(16, 16, 4), blk, 0, stream>>>(
        W2, 0, 512, 1,  bufB, 512 * L, (int)L, 1,
        bufD, 1024 * L, 1024, (int)L, 512, b2, invsig + 1, 1, nullptr, 0);
    // 5) h3 = lrelu((W3 h2)/s3 + b3)   [2048 x 2048]
    gemm_wmma_f32<<<dim3(16, 32, 4), blk, 0, stream>>>(
        W3, 0, 1024, 1, bufD, 1024 * L, (int)L, 1,
        bufA, 2048 * L, 2048, (int)L, 1024, b3, invsig + 2, 1, nullptr, 0);

    // ---- attention 1 ----
    gemm_wmma_f32<<<dim3(16, 4, 4), blk, 0, stream>>>(           // q1
        Wq1, 0, 2048, 1, bufA, 2048 * L, (int)L, 1,
        qbuf, 256 * L, 256, (int)L, 2048, bq1, nullptr, 0, nullptr, 0);
    gemm_wmma_f32<<<dim3(16, 4, 4), blk, 0, stream>>>(           // k1
        Wk1, 0, 2048, 1, bufA, 2048 * L, (int)L, 1,
        kbuf, 256 * L, 256, (int)L, 2048, bk1, nullptr, 0, nullptr, 0);
    gemm_wmma_f32<<<dim3(16, 32, 4), blk, 0, stream>>>(          // v1 (overwrites h1)
        Wv1, 0, 2048, 1, bufA, 2048 * L, (int)L, 1,
        bufB, 2048 * L, 2048, (int)L, 2048, bv1, nullptr, 0, nullptr, 0);
    // energy1[i][j] = sum_d q[d][i] k[d][j], masked -> p1
    gemm_wmma_f32<<<dim3(16, 32, 4), blk, 0, stream>>>(
        qbuf, 256 * L, 1, (int)L,  kbuf, 256 * L, (int)L, 1,
        p1, L * L, 2048, (int)L, 256, nullptr, nullptr, 0, dlen, 1);
    softmax_rows<<<dim3((unsigned)(BCONST * L)), dim3(256), 0, stream>>>(p1, (int)L);
    // h4[c][i] = sum_j v1[c][j] attn1[i][j]  (overwrites h3)
    gemm_wmma_f32<<<dim3(16, 32, 4), blk, 0, stream>>>(
        bufB, 2048 * L, (int)L, 1,  p1, L * L, 1, (int)L,
        bufA, 2048 * L, 2048, (int)L, 2048, nullptr, nullptr, 0, nullptr, 0);

    // ---- attention 2 ----
    gemm_wmma_f32<<<dim3(16, 4, 4), blk, 0, stream>>>(           // q2
        Wq2, 0, 2048, 1, bufA, 2048 * L, (int)L, 1,
        qbuf, 256 * L, 256, (int)L, 2048, bq2, nullptr, 0, nullptr, 0);
    gemm_wmma_f32<<<dim3(16, 4, 4), blk, 0, stream>>>(           // k2
        Wk2, 0, 2048, 1, bufA, 2048 * L, (int)L, 1,
        kbuf, 256 * L, 256, (int)L, 2048, bk2, nullptr, 0, nullptr, 0);
    gemm_wmma_f32<<<dim3(16, 32, 4), blk, 0, stream>>>(          // v2 (overwrites h2)
        Wv2, 0, 2048, 1, bufA, 2048 * L, (int)L, 1,
        bufD, 2048 * L, 2048, (int)L, 2048, bv2, nullptr, 0, nullptr, 0);
    gemm_wmma_f32<<<dim3(16, 32, 4), blk, 0, stream>>>(          // energy2 -> p2
        qbuf, 256 * L, 1, (int)L,  kbuf, 256 * L, (int)L, 1,
        p2, L * L, 2048, (int)L, 256, nullptr, nullptr, 0, dlen, 1);
    softmax_rows<<<dim3((unsigned)(BCONST * L)), dim3(256), 0, stream>>>(p2, (int)L);
    gemm_wmma_f32<<<dim3(16, 32, 4), blk, 0, stream>>>(          // h5 (overwrites v1)
        bufD, 2048 * L, (int)L, 1,  p2, L * L, 1, (int)L,
        bufB, 2048 * L, 2048, (int)L, 2048, nullptr, nullptr, 0, nullptr, 0);

    // ---- final conv + masked mean ----
    final_reduce<<<dim3(4), dim3(256), 0, stream>>>(bufB, Wl, bl, dlen, out);
}